// TemporalAttention_56959856279918
// MI455X (gfx1250) — compile-verified
//
#include <hip/hip_runtime.h>

typedef __attribute__((ext_vector_type(2))) float v2f;
typedef __attribute__((ext_vector_type(8))) float v8f;

#define TOKENS 8192     // N = H*W*D per batch
#define TDIM 12         // T
#define BATCH 2
#define TILE 16
#define WAVES_PER_BLOCK 8

// Prepass: per-row L2 norm and reciprocal (1/(norm+eps)) so the hot loop has no division.
__global__ __launch_bounds__(256)
void ta_row_norms_kernel(const float* __restrict__ q, float* __restrict__ norms,
                         float* __restrict__ rnorms, int total) {
    int i = blockIdx.x * blockDim.x + threadIdx.x;
    if (i < total) {
        const float* r = q + (size_t)i * TDIM;
        float s = 0.f;
#pragma unroll
        for (int t = 0; t < TDIM; ++t) s += r[t] * r[t];
        float n = sqrtf(s);
        norms[i]  = n;
        rnorms[i] = 1.0f / (n + 1e-6f);   // exact div once per row (prepass only)
    }
}

__global__ __launch_bounds__(256)
void ta_attn_kernel(const float* __restrict__ Q,
                    const float* __restrict__ V,
                    const float* __restrict__ norms,
                    const float* __restrict__ rnorms,
                    float* __restrict__ out) {
    const int lane = threadIdx.x & 31;
    const int widx = threadIdx.x >> 5;
    const int hi   = lane >> 4;   // 0/1 half of wave32
    const int lo   = lane & 15;
    const int lov  = (lo < TDIM) ? lo : 0;   // clamped V column (loop-invariant, in-bounds)

    // key-tile row permutation: perm = [0,1,4,5,8,9,12,13, 2,3,6,7,10,11,14,15]
    // With A rows loaded as key perm(lo), the S^T result registers (r = 2c+j) are
    // ALREADY in P's A-layout order for the PV matmul -> zero-cost transpose.
    const int pl = ((lo & 6) << 1) + (lo & 1) + ((lo >> 3) << 1);

    const int w  = blockIdx.x * WAVES_PER_BLOCK + widx;   // global wave id
    const int b  = w >> 9;                                // / (8192/16)
    const int n0 = (w & 511) << 4;                        // query tile base

    const float inv_sdk = 0.011048543456039806f;          // 1/sqrt(8192)
    const float L2E     = 1.4426950408889634f;
    const float l2s     = inv_sdk * L2E;                  // score scale pre-folded with log2(e)

    const float* __restrict__ q   = Q      + (size_t)b * TOKENS * TDIM;
    const float* __restrict__ v   = V      + (size_t)b * TOKENS * TDIM;
    const float* __restrict__ nb  = norms  + (size_t)b * TOKENS;
    const float* __restrict__ rnb = rnorms + (size_t)b * TOKENS;
    float*       __restrict__ o   = out    + (size_t)b * TOKENS * TDIM;

    // B operand (loop-invariant): query tile, lane holds B[4c+2hi+j][lo] = q[n0+lo][4c+2hi+j]
    const size_t qoff = (size_t)(n0 + lo) * TDIM + 2 * hi;
    v2f bq[3];
#pragma unroll
    for (int c = 0; c < 3; ++c)
        bq[c] = *(const v2f*)(q + qoff + 4 * c);

    // per-query softmax max bound (Cauchy-Schwarz: ||q[n]||/sdk >= all scores of row n),
    // pre-scaled by log2(e); one scalar per lane in the S^T layout (column = query lo)
    const float mq = nb[n0 + lo] * l2s;

    v8f oacc = {};        // O accumulator, C layout: [query r+8hi][t=lo], cols 0..11 valid
    v2f rs0 = {}, rs1 = {};   // paired rowsum accumulators (2 independent v_pk_add chains)

    const size_t koff = (size_t)pl * TDIM + 2 * hi;   // invariant lane offset, permuted keys

#pragma unroll 2
    for (int m0 = 0; m0 < TOKENS; m0 += TILE) {
        const float* qt = q + (size_t)m0 * TDIM;      // uniform tile bases
        const float* vt = v + (size_t)m0 * TDIM;

        // S^T = K_tile(permuted rows) x Q^T : A = normalized keys, B = queries
        const float rnk = rnb[m0 + pl];
        v8f st = {};
#pragma unroll
        for (int c = 0; c < 3; ++c) {
            v2f ak = *(const v2f*)(qt + koff + 4 * c);
            ak[0] *= rnk; ak[1] *= rnk;
            st = __builtin_amdgcn_wmma_f32_16x16x4_f32(false, ak, false, bq[c],
                                                       (short)0, st, false, false);
        }

        // softmax numerators: lane (hi,lo) reg r = P[query lo][key perm(r+8hi)]
        float pr[8];
#pragma unroll
        for (int r = 0; r < 8; ++r)
            pr[r] = __builtin_amdgcn_exp2f(st[r] * l2s - mq);

        // rowsum: 4 packed adds, two independent chains (no serialized scalar chain)
        v2f p01; p01[0] = pr[0]; p01[1] = pr[1];
        v2f p23; p23[0] = pr[2]; p23[1] = pr[3];
        v2f p45; p45[0] = pr[4]; p45[1] = pr[5];
        v2f p67; p67[0] = pr[6]; p67[1] = pr[7];
        rs0 += p01; rs1 += p23; rs0 += p45; rs1 += p67;

        // O += P x V_tile: pr pairs ARE the A-layout chunks (perm chosen so that
        // reg 2c+j holds key 4c+2hi+j); V rows load in natural order.
#pragma unroll
        for (int c = 0; c < 4; ++c) {
            const int k0 = 4 * c + 2 * hi;
            v2f pa; pa[0] = pr[2 * c]; pa[1] = pr[2 * c + 1];
            v2f bv;
            bv[0] = vt[(size_t)k0 * TDIM + lov];
            bv[1] = vt[(size_t)(k0 + 1) * TDIM + lov];
            oacc = __builtin_amdgcn_wmma_f32_16x16x4_f32(false, pa, false, bv,
                                                         (short)0, oacc, false, false);
        }
    }

    // total rowsum for query lo: fold pairs, then combine the two key-halves
    float rsacc = (rs0[0] + rs0[1]) + (rs1[0] + rs1[1]);
    rsacc += __shfl_xor(rsacc, 16);

    // normalize + diagonal replacement: attn[n,n] -> 0.0005 (applied post-softmax)
#pragma unroll
    for (int r = 0; r < 8; ++r) {
        const int n = n0 + r + 8 * hi;
        // broadcast rowsum of query r+8hi (held by lane index r+8hi) to this lane
        float sum_n   = __shfl(rsacc, r + 8 * hi);
        float inv_sum = __builtin_amdgcn_rcpf(sum_n);
        float nn = nb[n];
        // s[n,n]*log2e = nn^2/(nn+eps)*l2s via precomputed reciprocal; minus bound nn*l2s
        float pdiag = __builtin_amdgcn_exp2f(nn * nn * rnb[n] * l2s - nn * l2s);
        if (lo < TDIM) {
            float vd = v[(size_t)n * TDIM + lo];
            o[(size_t)n * TDIM + lo] =
                oacc[r] * inv_sum + (0.0005f - pdiag * inv_sum) * vd;
        }
    }
}

extern "C" void kernel_launch(void* const* d_in, const int* in_sizes, int n_in,
                              void* d_out, int out_size, void* d_ws, size_t ws_size,
                              hipStream_t stream) {
    const float* q = (const float*)d_in[0];
    const float* v = (const float*)d_in[1];
    float* norms  = (float*)d_ws;                    // BATCH*TOKENS floats
    float* rnorms = norms + BATCH * TOKENS;          // BATCH*TOKENS floats (128 KB total)
    float* out = (float*)d_out;

    int total = BATCH * TOKENS;
    ta_row_norms_kernel<<<(total + 255) / 256, 256, 0, stream>>>(q, norms, rnorms, total);

    int nwaves = (BATCH * TOKENS) / TILE;            // 1024 query tiles, 1 wave each
    ta_attn_kernel<<<nwaves / WAVES_PER_BLOCK, 256, 0, stream>>>(q, v, norms, rnorms, out);
}